// DifferentiableLindbladSimulator_19696720020125
// MI455X (gfx1250) — compile-verified
//
#include <hip/hip_runtime.h>

// Lindblad master-equation RK4 integrator for MI455X (gfx1250).
// One persistent workgroup (512 threads = 16 wave32s), state in LDS,
// all 64x64 complex GEMMs done with V_WMMA_F32_16X16X4_F32 tiles.

#define DD      64
#define PITCH   66            // LDS row pitch (floats) to avoid bank conflicts
#define NCTRL   4
#define NLOPS   4
#define NSEGS   200
#define NTHREADS 512
#define NELEM   (DD*DD)
#define MAT_FLOATS (DD*PITCH)

typedef float v2f __attribute__((ext_vector_type(2)));
typedef float v8f __attribute__((ext_vector_type(8)));

struct CMat {
    const float* re;   // real-part base pointer
    const float* im;   // imag-part base pointer
    int   s;           // element stride: 1 = planar (LDS), 2 = interleaved complex (global)
    int   p;           // row pitch in elements
    bool  trans;       // use M^T
    bool  conj;        // conjugate
};

// ---- WMMA f32 16x16x4 fragment loaders -------------------------------------
// A fragment: rows r0..r0+15, cols ka..ka+3.
// Layout: lanes 0-15 hold K=ka,ka+1; lanes 16-31 hold K=ka+2,ka+3 (2 VGPRs).
__device__ __forceinline__ v2f load_a(const float* P, int s, int p, bool trans,
                                      int r0, int ka, int lane) {
    int r = r0 + (lane & 15);
    int c = ka + ((lane >> 4) << 1);
    v2f f;
    if (!trans) {
        int i0 = r * p + c;
        f.x = P[s * i0];
        f.y = P[s * (i0 + 1)];
    } else {
        f.x = P[s * (c * p + r)];
        f.y = P[s * ((c + 1) * p + r)];
    }
    return f;
}

// B fragment: rows k0..k0+3, cols c0..c0+15.
// Layout: VGPR v, lanes 0-15 -> K=k0+v ; lanes 16-31 -> K=k0+v+2.
__device__ __forceinline__ v2f load_b(const float* P, int s, int p, bool trans,
                                      int k0, int c0, int lane) {
    int n = c0 + (lane & 15);
    int k = k0 + ((lane >> 4) << 1);
    v2f f;
    if (!trans) {
        f.x = P[s * (k * p + n)];
        f.y = P[s * ((k + 1) * p + n)];
    } else {
        f.x = P[s * (n * p + k)];
        f.y = P[s * (n * p + k + 1)];
    }
    return f;
}

__device__ __forceinline__ void wmma4(v8f& C, v2f a, v2f b) {
    // D = A(16x4) x B(4x16) + C, fp32 matrix pipe
    C = __builtin_amdgcn_wmma_f32_16x16x4_f32(false, a, false, b, (short)0, C,
                                              false, false);
}

// (Cr + i*Ci) += (ar + i*ai) * (A x B)   for this wave's 16x16 tile.
// f32 WMMA has no A/B NEG bits (ISA 7.12: NEG = CNeg only), so signs and the
// scalar prefactor are folded into the A fragments with VALU before the WMMA.
__device__ __forceinline__ void cgemm_tile(v8f& Cr, v8f& Ci, CMat A, CMat B,
                                           int rowT, int colT,
                                           float ar, float ai, int lane) {
    for (int kk = 0; kk < 4; ++kk) {
        #pragma unroll
        for (int kc = 0; kc < 4; ++kc) {
            const int ka = kk * 16 + kc * 4;
            v2f fAr = load_a(A.re, A.s, A.p, A.trans, rowT, ka, lane);
            v2f fAi = load_a(A.im, A.s, A.p, A.trans, rowT, ka, lane);
            v2f fBr = load_b(B.re, B.s, B.p, B.trans, ka, colT, lane);
            v2f fBi = load_b(B.im, B.s, B.p, B.trans, ka, colT, lane);
            if (A.conj) fAi = -fAi;
            if (B.conj) fBi = -fBi;
            if (ar != 0.0f) {                 // real prefactor component
                wmma4(Cr,  ar * fAr, fBr);
                wmma4(Cr, (-ar) * fAi, fBi);
                wmma4(Ci,  ar * fAr, fBi);
                wmma4(Ci,  ar * fAi, fBr);
            }
            if (ai != 0.0f) {                 // imaginary prefactor component
                wmma4(Cr, (-ai) * fAr, fBi);
                wmma4(Cr, (-ai) * fAi, fBr);
                wmma4(Ci,  ai * fAr, fBr);
                wmma4(Ci, (-ai) * fAi, fBi);
            }
        }
    }
}

// Store this wave's 16x16 f32 C/D tile (documented VGPR layout) to planar LDS.
__device__ __forceinline__ void store_tile(float* Pr, float* Pi,
                                           int rowT, int colT, int lane,
                                           v8f Cr, v8f Ci) {
    int half = lane >> 4;
    int col  = colT + (lane & 15);
    #pragma unroll
    for (int v = 0; v < 8; ++v) {
        int row = rowT + v + 8 * half;
        Pr[row * PITCH + col] = Cr[v];
        Pi[row * PITCH + col] = Ci[v];
    }
}

__device__ __forceinline__ bool finite_f(float x) {
    return fabsf(x) <= 3.4028234663852886e38f;   // false for NaN and Inf
}

__global__ __launch_bounds__(NTHREADS)
void lindblad_rk4_kernel(const float* __restrict__ rho0_ri,   // [2,64,64] planar
                         const float* __restrict__ cs,        // [200,4]
                         const float* __restrict__ H0,        // complex64 interleaved
                         const float* __restrict__ Hc,        // [4] complex64 interleaved
                         const float* __restrict__ Lops,      // [4] complex64 interleaved
                         const int*   __restrict__ Tptr,      // scalar int
                         float* __restrict__ out) {           // [2,64,64] planar
    extern __shared__ float smem[];
    float* sHr = smem;                       // Hamiltonian (per segment)
    float* sHi = sHr + MAT_FLOATS;
    float* sAr = sHi + MAT_FLOATS;           // Asum = sum_n Ln^dag Ln
    float* sAi = sAr + MAT_FLOATS;
    float* sRr = sAi + MAT_FLOATS;           // RK stage input rho
    float* sRi = sRr + MAT_FLOATS;
    float* sKr = sRi + MAT_FLOATS;           // Lindbladian output k
    float* sKi = sKr + MAT_FLOATS;
    float* sTr = sKi + MAT_FLOATS;           // temp: Ln @ rho
    float* sTi = sTr + MAT_FLOATS;
    float* sBr = sTi + MAT_FLOATS;           // base rho (step start / committed)
    float* sBi = sBr + MAT_FLOATS;
    float* sCr = sBi + MAT_FLOATS;           // RK4 accumulator
    float* sCi = sCr + MAT_FLOATS;
    float* sScal = sCi + MAT_FLOATS;         // [0]=scale,[1]=trace,[2]=final inv
    int*   sFlag = (int*)(sScal + 8);        // [0]=done, [1]=nonfinite

    const int tid  = threadIdx.x;
    const int lane = tid & 31;
    const int wave = tid >> 5;
    const int rowT = (wave >> 2) * 16;       // wave's output tile row
    const int colT = (wave & 3) * 16;        // wave's output tile col

    // Hint the constant operands toward the caches (global_prefetch_b8).
    for (int i = tid * 32; i < 2 * NELEM; i += NTHREADS * 32)
        __builtin_prefetch(H0 + i, 0, 2);
    for (int i = tid * 32; i < NCTRL * 2 * NELEM; i += NTHREADS * 32)
        __builtin_prefetch(Hc + i, 0, 2);
    for (int i = tid * 32; i < NLOPS * 2 * NELEM; i += NTHREADS * 32)
        __builtin_prefetch(Lops + i, 0, 2);

    // Integrator sizing (T is a runtime scalar input).
    const int Tint = Tptr[0];
    const double tseg = (double)Tint / (double)NSEGS;
    int nsub = (int)(tseg / 0.005);
    if (nsub < 1) nsub = 1;
    const float dtf = (float)(tseg / (double)nsub);

    // ---- Asum = sum_n Ln^dag Ln (WMMA, once) -------------------------------
    {
        v8f Cr = {0.f,0.f,0.f,0.f,0.f,0.f,0.f,0.f};
        v8f Ci = {0.f,0.f,0.f,0.f,0.f,0.f,0.f,0.f};
        for (int n = 0; n < NLOPS; ++n) {
            const float* Lb = Lops + n * 2 * NELEM;
            CMat Adag{Lb, Lb + 1, 2, DD, true,  true };   // Ln^dag
            CMat Bm  {Lb, Lb + 1, 2, DD, false, false};   // Ln
            cgemm_tile(Cr, Ci, Adag, Bm, rowT, colT, 1.0f, 0.0f, lane);
        }
        store_tile(sAr, sAi, rowT, colT, lane, Cr, Ci);
    }

    // ---- initial rho with trace fixup --------------------------------------
    if (tid == 0) {
        float tr = 0.f;
        for (int i = 0; i < DD; ++i) tr += rho0_ri[i * DD + i];
        float sc = 1.0f;
        if (fabsf(tr - 1.0f) > 0.01f && tr > 1e-10f) sc = 1.0f / tr;
        sScal[0] = sc;
        sFlag[0] = 0;
    }
    __syncthreads();
    {
        const float sc0 = sScal[0];
        for (int e = tid; e < NELEM; e += NTHREADS) {
            int r = e >> 6, c = e & 63, li = r * PITCH + c;
            sBr[li] = rho0_ri[e] * sc0;
            sBi[li] = rho0_ri[NELEM + e] * sc0;
        }
    }
    __syncthreads();

    const float cin[4] = {0.f, 0.5f, 0.5f, 1.f};          // * dt applied below
    const float wgt[4] = {1.f/6.f, 1.f/3.f, 1.f/3.f, 1.f/6.f};

    // ======================= time evolution =================================
    for (int seg = 0; seg < NSEGS; ++seg) {
        // clamped control amplitudes (uniform scalars)
        float u[NCTRL];
        #pragma unroll
        for (int k = 0; k < NCTRL; ++k) {
            float v = cs[seg * NCTRL + k];
            u[k] = fminf(fmaxf(v, -10.0f), 10.0f);
        }
        // H = H0 + sum_k u_k Hc_k   (elementwise, into LDS)
        for (int e = tid; e < NELEM; e += NTHREADS) {
            int r = e >> 6, c = e & 63, li = r * PITCH + c;
            float hr = H0[2 * e], hi = H0[2 * e + 1];
            #pragma unroll
            for (int k = 0; k < NCTRL; ++k) {
                hr += u[k] * Hc[k * 2 * NELEM + 2 * e];
                hi += u[k] * Hc[k * 2 * NELEM + 2 * e + 1];
            }
            sHr[li] = hr; sHi[li] = hi;
        }
        __syncthreads();

        for (int sub = 0; sub < nsub; ++sub) {
            const int doneFlag = sFlag[0];       // uniform across the block
            if (doneFlag) continue;              // frozen after early return

            // acc = base
            for (int e = tid; e < NELEM; e += NTHREADS) {
                int r = e >> 6, c = e & 63, li = r * PITCH + c;
                sCr[li] = sBr[li]; sCi[li] = sBi[li];
            }
            __syncthreads();

            for (int st = 0; st < 4; ++st) {
                const float* Sr; const float* Si;
                if (st == 0) { Sr = sBr; Si = sBi; }
                else {
                    const float cc = cin[st] * dtf;
                    for (int e = tid; e < NELEM; e += NTHREADS) {
                        int r = e >> 6, c = e & 63, li = r * PITCH + c;
                        sRr[li] = sBr[li] + cc * sKr[li];
                        sRi[li] = sBi[li] + cc * sKi[li];
                    }
                    __syncthreads();
                    Sr = sRr; Si = sRi;
                }

                // ---- k = -i[H,rho] + sum_n Ln rho Ln^dag - 0.5{Asum,rho} ---
                v8f Kr = {0.f,0.f,0.f,0.f,0.f,0.f,0.f,0.f};
                v8f Ki = {0.f,0.f,0.f,0.f,0.f,0.f,0.f,0.f};
                CMat S {Sr,  Si,  1, PITCH, false, false};
                CMat Hm{sHr, sHi, 1, PITCH, false, false};
                CMat Am{sAr, sAi, 1, PITCH, false, false};
                cgemm_tile(Kr, Ki, Hm, S, rowT, colT, 0.0f, -1.0f, lane); // -i H rho
                cgemm_tile(Kr, Ki, S, Hm, rowT, colT, 0.0f, +1.0f, lane); // +i rho H
                cgemm_tile(Kr, Ki, Am, S, rowT, colT, -0.5f, 0.0f, lane);
                cgemm_tile(Kr, Ki, S, Am, rowT, colT, -0.5f, 0.0f, lane);
                for (int n = 0; n < NLOPS; ++n) {
                    const float* Lb = Lops + n * 2 * NELEM;
                    CMat Ln{Lb, Lb + 1, 2, DD, false, false};
                    v8f Tr2 = {0.f,0.f,0.f,0.f,0.f,0.f,0.f,0.f};
                    v8f Ti2 = {0.f,0.f,0.f,0.f,0.f,0.f,0.f,0.f};
                    cgemm_tile(Tr2, Ti2, Ln, S, rowT, colT, 1.0f, 0.0f, lane);
                    store_tile(sTr, sTi, rowT, colT, lane, Tr2, Ti2);
                    __syncthreads();             // whole Ln@rho visible
                    CMat Tm  {sTr, sTi, 1, PITCH, false, false};
                    CMat Ldag{Lb, Lb + 1, 2, DD, true, true};
                    cgemm_tile(Kr, Ki, Tm, Ldag, rowT, colT, 1.0f, 0.0f, lane);
                    __syncthreads();             // before sT reuse
                }
                store_tile(sKr, sKi, rowT, colT, lane, Kr, Ki);
                __syncthreads();

                const float w = wgt[st] * dtf;
                for (int e = tid; e < NELEM; e += NTHREADS) {
                    int r = e >> 6, c = e & 63, li = r * PITCH + c;
                    sCr[li] += w * sKr[li];
                    sCi[li] += w * sKi[li];
                }
                __syncthreads();
            }

            // ---- commit: trace normalize / early-return / NaN reset --------
            if (tid == 0) sFlag[1] = 0;
            __syncthreads();
            int localBad = 0;
            for (int e = tid; e < NELEM; e += NTHREADS) {
                int r = e >> 6, c = e & 63, li = r * PITCH + c;
                if (!finite_f(sCr[li]) || !finite_f(sCi[li])) localBad = 1;
            }
            if (localBad) atomicOr(&sFlag[1], 1);
            if (tid == 0) {
                float tr = 0.f;
                for (int i = 0; i < DD; ++i) tr += sCr[i * PITCH + i];
                sScal[1] = tr;
            }
            __syncthreads();
            const float tr   = sScal[1];
            const int nonfin = sFlag[1];
            const bool badTr = (tr <= 1e-10f);
            if (badTr || nonfin) {               // reset to raw rho0
                for (int e = tid; e < NELEM; e += NTHREADS) {
                    int r = e >> 6, c = e & 63, li = r * PITCH + c;
                    sBr[li] = rho0_ri[e];
                    sBi[li] = rho0_ri[NELEM + e];
                }
                if (tid == 0 && badTr) sFlag[0] = 1;   // freeze evolution
            } else {
                const float inv = 1.0f / tr;
                for (int e = tid; e < NELEM; e += NTHREADS) {
                    int r = e >> 6, c = e & 63, li = r * PITCH + c;
                    sBr[li] = sCr[li] * inv;
                    sBi[li] = sCi[li] * inv;
                }
            }
            __syncthreads();
        }
    }

    // ---- final normalize + write out [2,64,64] -----------------------------
    if (tid == 0) {
        float tr = 0.f;
        for (int i = 0; i < DD; ++i) tr += sBr[i * PITCH + i];
        sScal[2] = (tr > 1e-10f) ? 1.0f / tr : 1.0f;
    }
    __syncthreads();
    const float invf = sScal[2];
    for (int e = tid; e < NELEM; e += NTHREADS) {
        int r = e >> 6, c = e & 63, li = r * PITCH + c;
        out[e]         = sBr[li] * invf;
        out[NELEM + e] = sBi[li] * invf;
    }
}

extern "C" void kernel_launch(void* const* d_in, const int* in_sizes, int n_in,
                              void* d_out, int out_size, void* d_ws, size_t ws_size,
                              hipStream_t stream) {
    (void)in_sizes; (void)n_in; (void)out_size; (void)d_ws; (void)ws_size;
    const float* rho0 = (const float*)d_in[0];   // [2,64,64] f32
    const float* cs   = (const float*)d_in[1];   // [200,4]   f32
    const float* H0   = (const float*)d_in[2];   // complex64 interleaved
    const float* Hc   = (const float*)d_in[3];   // [4] complex64 interleaved
    const float* L    = (const float*)d_in[4];   // [4] complex64 interleaved
    const int*   T    = (const int*)d_in[5];     // scalar
    float* out = (float*)d_out;

    const size_t smem = (size_t)(14 * MAT_FLOATS + 16) * sizeof(float); // ~237 KB
    hipLaunchKernelGGL(lindblad_rk4_kernel, dim3(1), dim3(NTHREADS), smem, stream,
                       rho0, cs, H0, Hc, L, T, out);
}